// SyncSwitchTransformersSparseMLP_72610717106851
// MI455X (gfx1250) — compile-verified
//
#include <hip/hip_runtime.h>
#include <hip/hip_bf16.h>
#include <math.h>

// Problem constants (from reference): B=2,S=512,D=1024,F=10240,E=8, fp32.
#define Bv 2
#define Sv 512
#define Tv (Bv * Sv)   // 1024 tokens
#define Dv 1024
#define Fv 10240
#define Ev 8

#define NTILE 512          // columns per block tile (8 waves x 64)
#define KT 16              // K tile per LDS stage
#define BPAD 16            // LDS pad: (528-16)%64==0 -> lane halves on disjoint banks
#define BSTRIDE (NTILE + BPAD)

typedef float v2f __attribute__((ext_vector_type(2)));
typedef float v8f __attribute__((ext_vector_type(8)));
typedef int v4i __attribute__((vector_size(4 * sizeof(int))));

// ---- CDNA5 async global->LDS copy (ASYNCcnt path), with safe fallback ----
#if defined(__has_builtin)
#if __has_builtin(__builtin_amdgcn_global_load_async_to_lds_b128)
#define USE_ASYNC_LDS 1
#endif
#endif
#ifndef USE_ASYNC_LDS
#define USE_ASYNC_LDS 0
#endif

__device__ __forceinline__ void copy16_to_lds(const float* g, float* l) {
#if USE_ASYNC_LDS
    // global_load_async_to_lds_b128: per-lane 16B global -> LDS, ASYNCcnt-tracked
    __builtin_amdgcn_global_load_async_to_lds_b128(
        (v4i*)(void*)const_cast<float*>(g), (v4i*)(void*)l, 0, 0);
#else
    *reinterpret_cast<float4*>(l) = *reinterpret_cast<const float4*>(g);
#endif
}

__device__ __forceinline__ void lds_copy_join() {
#if USE_ASYNC_LDS
#if __has_builtin(__builtin_amdgcn_s_wait_asynccnt)
    __builtin_amdgcn_s_wait_asynccnt(0);
#else
    asm volatile("s_wait_asynccnt 0x0" ::: "memory");
#endif
#endif
}

__device__ __forceinline__ float gelu_tanh(float x) {
    const float c0 = 0.79788456080286535588f; // sqrt(2/pi)
    float x3 = x * x * x;
    return 0.5f * x * (1.0f + tanhf(c0 * (x + 0.044715f * x3)));
}

__global__ void init_counts_kernel(int* __restrict__ counts) {
    if (threadIdx.x < Ev) counts[threadIdx.x] = 0;
}

// One block (256 threads) per token: router logits, softmax top-1 prob,
// argmax, and compacted per-expert token lists via atomics.
__global__ __launch_bounds__(256)
void router_kernel(const float* __restrict__ hs, const float* __restrict__ Wr,
                   float* __restrict__ logits, int* __restrict__ eidx,
                   float* __restrict__ probs, int* __restrict__ counts,
                   int* __restrict__ perm) {
    const int t = blockIdx.x;
    const int tid = threadIdx.x;

    float acc[Ev];
#pragma unroll
    for (int e = 0; e < Ev; ++e) acc[e] = 0.0f;

    const float* x = hs + (size_t)t * Dv;
#pragma unroll
    for (int i = 0; i < Dv / 256; ++i) {
        const int d = tid + i * 256;
        const float xv = x[d];
#pragma unroll
        for (int e = 0; e < Ev; ++e) acc[e] += xv * Wr[d * Ev + e];
    }

    __shared__ float red[256][Ev];
#pragma unroll
    for (int e = 0; e < Ev; ++e) red[tid][e] = acc[e];
    __syncthreads();
    for (int s = 128; s > 0; s >>= 1) {
        if (tid < s) {
#pragma unroll
            for (int e = 0; e < Ev; ++e) red[tid][e] += red[tid + s][e];
        }
        __syncthreads();
    }

    if (tid == 0) {
        float mx = -INFINITY;
        int am = 0;
#pragma unroll
        for (int e = 0; e < Ev; ++e) {
            const float lg = red[0][e];
            logits[(size_t)t * Ev + e] = lg;
            if (lg > mx) { mx = lg; am = e; }  // strict '>' matches jnp.argmax tie rule
        }
        float denom = 0.0f;
#pragma unroll
        for (int e = 0; e < Ev; ++e) denom += __expf(red[0][e] - mx);
        probs[t] = 1.0f / denom;   // softmax value at the argmax
        eidx[t] = am;
        const int slot = atomicAdd(&counts[am], 1);
        perm[am * Tv + slot] = t;  // order nondeterministic, results order-invariant
    }
}

// Up-projection + GEGLU: H[t, :] = gelu(x @ W0[e]) * (x @ W1[e]) for tokens
// routed to expert e. Block tile: 16 tokens x 512 cols; 8 waves, each wave
// owns 64 cols with dual v8f accumulators (W0 path and W1 path).
__global__ __launch_bounds__(256)
void expert_up_kernel(const float* __restrict__ hs,
                      const float* __restrict__ W0g, const float* __restrict__ W1g,
                      const int* __restrict__ counts, const int* __restrict__ perm,
                      float* __restrict__ H) {
    const int e = blockIdx.z;
    const int cnt = counts[e];
    const int mt = blockIdx.y;
    if (mt * 16 >= cnt) return;
    const int nbase = blockIdx.x * NTILE;
    const int tid = threadIdx.x;
    const int wave = tid >> 5;
    const int lane = tid & 31;

    __shared__ float As[16][KT + 1];
    __shared__ float Bs0[KT][BSTRIDE];
    __shared__ float Bs1[KT][BSTRIDE];
    __shared__ int toks[16];

    if (tid < 16) {
        const int idx = mt * 16 + tid;
        toks[tid] = (idx < cnt) ? perm[e * Tv + idx] : perm[e * Tv + mt * 16];
    }
    __syncthreads();
    const int am = tid >> 4;   // A-stage row this thread loads
    const int ak = tid & 15;   // A-stage col this thread loads
    const size_t arow = (size_t)toks[am] * Dv;

    v8f acc0[4] = {};
    v8f acc1[4] = {};

    const float* W0e = W0g + (size_t)e * Dv * Fv;
    const float* W1e = W1g + (size_t)e * Dv * Fv;

    const int mrow = lane & 15;          // A: lane -> M ; B: lane -> N
    const int khalf = (lane >> 4) << 1;  // lanes 16-31 carry K+2,K+3

    for (int k0 = 0; k0 < Dv; k0 += KT) {
        __syncthreads();
        As[am][ak] = hs[arow + k0 + ak];
#pragma unroll
        for (int i = 0; i < 8; ++i) {
            const int flat = (i * 256 + tid) * 4;
            const int kr = flat >> 9;        // /512
            const int nc = flat & 511;
            const size_t goff = (size_t)(k0 + kr) * Fv + nbase + nc;
            copy16_to_lds(&W0e[goff], &Bs0[kr][nc]);
            copy16_to_lds(&W1e[goff], &Bs1[kr][nc]);
        }
        lds_copy_join();
        __syncthreads();
        if (k0 + KT < Dv) {
            // Each of 256 threads hints one 128B line of the next tiles.
            const size_t p = (size_t)(k0 + KT + (tid >> 4)) * Fv + nbase + (tid & 15) * 32;
            __builtin_prefetch(&W0e[p], 0, 1);
            __builtin_prefetch(&W1e[p], 0, 1);
        }
#pragma unroll
        for (int kk = 0; kk < KT; kk += 4) {
            v2f a;
            a.x = As[mrow][kk + khalf];
            a.y = As[mrow][kk + khalf + 1];
#pragma unroll
            for (int j = 0; j < 4; ++j) {
                const int n = wave * 64 + j * 16 + mrow;
                v2f b0, b1;
                b0.x = Bs0[kk + khalf][n];
                b0.y = Bs0[kk + khalf + 1][n];
                b1.x = Bs1[kk + khalf][n];
                b1.y = Bs1[kk + khalf + 1][n];
                acc0[j] = __builtin_amdgcn_wmma_f32_16x16x4_f32(
                    false, a, false, b0, (short)0, acc0[j], false, false);
                acc1[j] = __builtin_amdgcn_wmma_f32_16x16x4_f32(
                    false, a, false, b1, (short)0, acc1[j], false, false);
            }
        }
    }

    // Epilogue: fused GEGLU, scatter rows back by original token id.
    const int rbase = (lane >> 4) << 3;  // C/D layout: M = r + 8*(lane>=16)
#pragma unroll
    for (int j = 0; j < 4; ++j) {
        const int n = nbase + wave * 64 + j * 16 + mrow;
#pragma unroll
        for (int r = 0; r < 8; ++r) {
            const int m = rbase + r;
            if (mt * 16 + m < cnt) {
                const float hv = gelu_tanh(acc0[j][r]) * acc1[j][r];
                H[(size_t)toks[m] * Fv + n] = hv;
            }
        }
    }
}

// Down-projection: out[t, :] = prob[t] * (H[t, :] @ Wo[e]) for routed tokens.
__global__ __launch_bounds__(256)
void expert_down_kernel(const float* __restrict__ H, const float* __restrict__ Wog,
                        const int* __restrict__ counts, const int* __restrict__ perm,
                        const float* __restrict__ probs, float* __restrict__ out) {
    const int e = blockIdx.z;
    const int cnt = counts[e];
    const int mt = blockIdx.y;
    if (mt * 16 >= cnt) return;
    const int nbase = blockIdx.x * NTILE;  // over Dv
    const int tid = threadIdx.x;
    const int wave = tid >> 5;
    const int lane = tid & 31;

    __shared__ float As[16][KT + 1];
    __shared__ float Bs[KT][BSTRIDE];
    __shared__ int toks[16];

    if (tid < 16) {
        const int idx = mt * 16 + tid;
        toks[tid] = (idx < cnt) ? perm[e * Tv + idx] : perm[e * Tv + mt * 16];
    }
    __syncthreads();
    const int am = tid >> 4;
    const int ak = tid & 15;
    const size_t arow = (size_t)toks[am] * Fv;

    v8f acc[4] = {};

    const float* Woe = Wog + (size_t)e * Fv * Dv;
    const int mrow = lane & 15;
    const int khalf = (lane >> 4) << 1;

    for (int k0 = 0; k0 < Fv; k0 += KT) {
        __syncthreads();
        As[am][ak] = H[arow + k0 + ak];
#pragma unroll
        for (int i = 0; i < 8; ++i) {
            const int flat = (i * 256 + tid) * 4;
            const int kr = flat >> 9;
            const int nc = flat & 511;
            copy16_to_lds(&Woe[(size_t)(k0 + kr) * Dv + nbase + nc], &Bs[kr][nc]);
        }
        lds_copy_join();
        __syncthreads();
        if (k0 + KT < Fv) {
            const size_t p = (size_t)(k0 + KT + (tid >> 4)) * Dv + nbase + (tid & 15) * 32;
            __builtin_prefetch(&Woe[p], 0, 1);
        }
#pragma unroll
        for (int kk = 0; kk < KT; kk += 4) {
            v2f a;
            a.x = As[mrow][kk + khalf];
            a.y = As[mrow][kk + khalf + 1];
#pragma unroll
            for (int j = 0; j < 4; ++j) {
                const int n = wave * 64 + j * 16 + mrow;
                v2f b;
                b.x = Bs[kk + khalf][n];
                b.y = Bs[kk + khalf + 1][n];
                acc[j] = __builtin_amdgcn_wmma_f32_16x16x4_f32(
                    false, a, false, b, (short)0, acc[j], false, false);
            }
        }
    }

    const int rbase = (lane >> 4) << 3;
#pragma unroll
    for (int j = 0; j < 4; ++j) {
        const int n = nbase + wave * 64 + j * 16 + mrow;
#pragma unroll
        for (int r = 0; r < 8; ++r) {
            const int m = rbase + r;
            if (mt * 16 + m < cnt) {
                const int t = toks[m];
                out[(size_t)t * Dv + n] = probs[t] * acc[j][r];
            }
        }
    }
}

extern "C" void kernel_launch(void* const* d_in, const int* in_sizes, int n_in,
                              void* d_out, int out_size, void* d_ws, size_t ws_size,
                              hipStream_t stream) {
    const float* hs = (const float*)d_in[0];  // [B,S,D]
    const float* Wr = (const float*)d_in[1];  // [D,E]
    const float* W0 = (const float*)d_in[2];  // [E,D,F]
    const float* W1 = (const float*)d_in[3];  // [E,D,F]
    const float* Wo = (const float*)d_in[4];  // [E,F,D]

    // d_out: out [T,D] fp32 | router_logits [T,E] fp32 | expert_index [T] int32
    float* out = (float*)d_out;
    float* logits = out + (size_t)Tv * Dv;
    int* eidx = (int*)(logits + (size_t)Tv * Ev);

    // Workspace: H [T,F] fp32 | probs [T] fp32 | counts [E] i32 | perm [E,T] i32
    float* ws = (float*)d_ws;
    float* H = ws;
    float* probs = ws + (size_t)Tv * Fv;
    int* counts = (int*)(probs + Tv);
    int* perm = counts + Ev;

    init_counts_kernel<<<1, 32, 0, stream>>>(counts);
    router_kernel<<<Tv, 256, 0, stream>>>(hs, Wr, logits, eidx, probs, counts, perm);

    dim3 gUp(Fv / NTILE, Tv / 16, Ev);   // 20 x 64 x 8 (empty tiles early-exit)
    expert_up_kernel<<<gUp, 256, 0, stream>>>(hs, W0, W1, counts, perm, H);

    dim3 gDown(Dv / NTILE, Tv / 16, Ev); // 2 x 64 x 8
    expert_down_kernel<<<gDown, 256, 0, stream>>>(H, Wo, counts, perm, probs, out);
}